// Att_Decoder_27049704030248
// MI455X (gfx1250) — compile-verified
//
#include <hip/hip_runtime.h>
#include <hip/hip_bf16.h>
#include <math.h>

// ---------------------------------------------------------------------------
// Att decoder step, MI455X (gfx1250, wave32).
// Bandwidth-bound GEMV chain (~264 MB fp32 weights/call -> ~11.3 us floor).
// GEMVs run through V_WMMA_F32_16X16X4_F32: A = 16x4 weight tile (streamed,
// no duplication), B = activation broadcast across N, D columns replicated.
// ---------------------------------------------------------------------------

typedef float v2f __attribute__((ext_vector_type(2)));
typedef float v8f __attribute__((ext_vector_type(8)));

#define H_DIM   1024
#define V_DIM   50000
#define ML_DIM  50
#define TWO_H   2048
#define THREE_H 3072

// workspace layout (float offsets)
#define WS_XA     0                         // 2048: [emb, att_applied]
#define WS_X0     (WS_XA + TWO_H)           // 1024: combine output
#define WS_G      (WS_X0 + H_DIM)           // 6144: [gi(3H) | gh(3H)]
#define WS_H0     (WS_G + 2*THREE_H)        // 1024
#define WS_H1     (WS_H0 + H_DIM)           // 1024
#define WS_LOGITS (WS_H1 + H_DIM)           // 50000
#define WS_SCAL   (WS_LOGITS + V_DIM)       // 2: max, log(sum exp)

// d_out layout: [log_softmax 50000 | new_hidden 2048 | att_w 50]
#define OUT_HID  V_DIM
#define OUT_ATTW (V_DIM + 2*H_DIM)

__device__ __forceinline__ float sigf(float x) { return 1.0f / (1.0f + __expf(-x)); }

// --------------------------------------------------------------------------
// Core: one wave computes y[row0..row0+15] = W[row0..row0+15][:] . x  (+bias)
// via V_WMMA_F32_16X16X4_F32. K must be a multiple of 8.
// A layout (32-bit, 16x4): lanes 0-15 -> M=lane, K={0,1}; lanes 16-31 -> K={2,3}.
// B layout mirrors the lane-half K split; x broadcast across N makes every
// D column identical: lane0 acc[r]=y[row0+r], lane16 acc[r]=y[row0+8+r].
// --------------------------------------------------------------------------
__device__ __forceinline__ void gemv16_wmma(const float* __restrict__ W,
                                            const float* __restrict__ x,
                                            const float* __restrict__ bias,
                                            float* __restrict__ y,
                                            int K, int row0, int relu)
{
    const int lane  = threadIdx.x & 31;
    const int m     = lane & 15;
    const int khalf = (lane >> 4) << 1;           // 0 or 2

    const float* __restrict__ wrow = W + (size_t)(row0 + m) * (size_t)K + khalf;
    const float* __restrict__ xk   = x + khalf;

    v8f acc0 = {};
    v8f acc1 = {};
    for (int kb = 0; kb < K; kb += 8) {
        v2f a0 = *(const v2f*)(wrow + kb);
        v2f b0 = *(const v2f*)(xk + kb);
        v2f a1 = *(const v2f*)(wrow + kb + 4);
        v2f b1 = *(const v2f*)(xk + kb + 4);
        acc0 = __builtin_amdgcn_wmma_f32_16x16x4_f32(false, a0, false, b0,
                                                     (short)0, acc0, false, false);
        acc1 = __builtin_amdgcn_wmma_f32_16x16x4_f32(false, a1, false, b1,
                                                     (short)0, acc1, false, false);
    }
    v8f acc = acc0 + acc1;

    if (lane == 0) {
        #pragma unroll
        for (int r = 0; r < 8; ++r) {
            float v = acc[r] + bias[row0 + r];
            y[row0 + r] = relu ? fmaxf(v, 0.0f) : v;
        }
    } else if (lane == 16) {
        #pragma unroll
        for (int r = 0; r < 8; ++r) {
            float v = acc[r] + bias[row0 + 8 + r];
            y[row0 + 8 + r] = relu ? fmaxf(v, 0.0f) : v;
        }
    }
}

// --------------------------------------------------------------------------
// K1: embedding gather, attention logits + softmax, attention apply,
//     build concat vector [emb, att_applied] for the combine GEMV.
// One block of 1024 threads (32 waves).
// --------------------------------------------------------------------------
__global__ void k_prologue(const int* __restrict__ input,
                           const float* __restrict__ hidden,
                           const float* __restrict__ enc,
                           const float* __restrict__ emb_W,
                           const float* __restrict__ attn_W,
                           const float* __restrict__ attn_b,
                           float* __restrict__ ws,
                           float* __restrict__ dout)
{
    __shared__ float s_eh[TWO_H];     // [emb, hidden0]
    __shared__ float s_logit[ML_DIM];
    __shared__ float s_attw[ML_DIM];

    const int t = threadIdx.x;        // 0..1023
    const int tok = input[0];

    float e = emb_W[(size_t)tok * H_DIM + t];
    s_eh[t]          = e;
    s_eh[H_DIM + t]  = hidden[t];     // hidden[0][0][:]
    ws[WS_XA + t]    = e;
    __syncthreads();

    // attention logits: 50 rows of length 2048, one wave per row
    const int w = t >> 5, lane = t & 31;
    for (int r = w; r < ML_DIM; r += 32) {
        float p = 0.0f;
        for (int j = lane; j < TWO_H; j += 32)
            p += attn_W[(size_t)r * TWO_H + j] * s_eh[j];
        #pragma unroll
        for (int off = 16; off; off >>= 1) p += __shfl_xor(p, off, 32);
        if (lane == 0) s_logit[r] = p + attn_b[r];
    }
    __syncthreads();

    // softmax over 50 (serial on thread 0: trivial)
    if (t == 0) {
        float mx = -INFINITY;
        for (int r = 0; r < ML_DIM; ++r) mx = fmaxf(mx, s_logit[r]);
        float s = 0.0f;
        for (int r = 0; r < ML_DIM; ++r) {
            float v = __expf(s_logit[r] - mx);
            s_attw[r] = v; s += v;
        }
        float inv = 1.0f / s;
        for (int r = 0; r < ML_DIM; ++r) {
            s_attw[r] *= inv;
            dout[OUT_ATTW + r] = s_attw[r];
        }
    }
    __syncthreads();

    // att_applied[t] = sum_s attw[s] * enc[s][t]
    float a = 0.0f;
    for (int s = 0; s < ML_DIM; ++s) a += s_attw[s] * enc[(size_t)s * H_DIM + t];
    ws[WS_XA + H_DIM + t] = a;
}

// --------------------------------------------------------------------------
// Generic WMMA GEMV kernel: nWaves waves, 16 rows each.
// --------------------------------------------------------------------------
__global__ void k_gemv16(const float* __restrict__ W, const float* __restrict__ x,
                         const float* __restrict__ bias, float* __restrict__ y,
                         int K, int nWaves, int relu)
{
    int wave = blockIdx.x * 8 + (threadIdx.x >> 5);
    if (wave >= nWaves) return;                     // whole-wave uniform exit
    gemv16_wmma(W, x, bias, y, K, wave << 4, relu);
}

// --------------------------------------------------------------------------
// GRU matmuls for one layer: g[0..3071] = Wih.x + bih ; g[3072..6143] = Whh.h + bhh
// 384 waves (48 blocks x 8 waves); a wave never straddles the 3072 boundary.
// --------------------------------------------------------------------------
__global__ void k_gru_gemv(const float* __restrict__ Wih, const float* __restrict__ Whh,
                           const float* __restrict__ bih, const float* __restrict__ bhh,
                           const float* __restrict__ xin, const float* __restrict__ h,
                           float* __restrict__ g)
{
    int wave = blockIdx.x * 8 + (threadIdx.x >> 5);  // 0..383
    if (wave < 192) {
        gemv16_wmma(Wih, xin, bih, g, H_DIM, wave << 4, 0);
    } else {
        gemv16_wmma(Whh, h, bhh, g + THREE_H, H_DIM, (wave - 192) << 4, 0);
    }
}

// --------------------------------------------------------------------------
// GRU gate nonlinearity (PyTorch order r,z,n). 1024 elements.
// --------------------------------------------------------------------------
__global__ void k_gru_gate(const float* __restrict__ g,
                           const float* __restrict__ h_prev,
                           float* __restrict__ h_new_ws,
                           float* __restrict__ dout_h)
{
    int j = blockIdx.x * 256 + threadIdx.x;
    if (j >= H_DIM) return;
    float r = sigf(g[j]            + g[THREE_H + j]);
    float z = sigf(g[H_DIM + j]    + g[THREE_H + H_DIM + j]);
    float n = tanhf(g[2*H_DIM + j] + r * g[THREE_H + 2*H_DIM + j]);
    float h = h_prev[j];
    float hn = (1.0f - z) * n + z * h;
    h_new_ws[j] = hn;
    dout_h[j]   = hn;
}

// --------------------------------------------------------------------------
// max + log-sum-exp over 50000 logits (one block, 1024 threads).
// --------------------------------------------------------------------------
__global__ void k_lse(const float* __restrict__ logits, float* __restrict__ scal)
{
    __shared__ float redm[32];
    __shared__ float reds[32];
    const int t = threadIdx.x, lane = t & 31, w = t >> 5;

    float m = -INFINITY;
    for (int i = t; i < V_DIM; i += 1024) m = fmaxf(m, logits[i]);
    #pragma unroll
    for (int off = 16; off; off >>= 1) m = fmaxf(m, __shfl_xor(m, off, 32));
    if (lane == 0) redm[w] = m;
    __syncthreads();
    float gmax;
    {
        float v = redm[lane];
        #pragma unroll
        for (int off = 16; off; off >>= 1) v = fmaxf(v, __shfl_xor(v, off, 32));
        gmax = v;   // every lane of every wave computes the same reduction
    }

    float s = 0.0f;
    for (int i = t; i < V_DIM; i += 1024) s += __expf(logits[i] - gmax);
    #pragma unroll
    for (int off = 16; off; off >>= 1) s += __shfl_xor(s, off, 32);
    if (lane == 0) reds[w] = s;
    __syncthreads();
    if (t == 0) {
        float tot = 0.0f;
        for (int i = 0; i < 32; ++i) tot += reds[i];
        scal[0] = gmax;
        scal[1] = logf(tot);
    }
}

__global__ void k_logsoftmax(const float* __restrict__ logits,
                             const float* __restrict__ scal,
                             float* __restrict__ out)
{
    int i = blockIdx.x * 256 + threadIdx.x;
    if (i < V_DIM) out[i] = logits[i] - scal[0] - scal[1];
}

// --------------------------------------------------------------------------
// launch
// --------------------------------------------------------------------------
extern "C" void kernel_launch(void* const* d_in, const int* in_sizes, int n_in,
                              void* d_out, int out_size, void* d_ws, size_t ws_size,
                              hipStream_t stream)
{
    const int*   input   = (const int*)  d_in[0];
    const float* hidden  = (const float*)d_in[1];   // (2,1,1024)
    const float* enc     = (const float*)d_in[2];   // (50,1024)
    const float* emb_W   = (const float*)d_in[3];   // (50000,1024)
    const float* attn_W  = (const float*)d_in[4];   // (50,2048)
    const float* attn_b  = (const float*)d_in[5];   // (50,)
    const float* comb_W  = (const float*)d_in[6];   // (1024,2048)
    const float* comb_b  = (const float*)d_in[7];   // (1024,)
    const float* gru_Wih = (const float*)d_in[8];   // (2,3072,1024)
    const float* gru_Whh = (const float*)d_in[9];   // (2,3072,1024)
    const float* gru_bih = (const float*)d_in[10];  // (2,3072)
    const float* gru_bhh = (const float*)d_in[11];  // (2,3072)
    const float* out_W   = (const float*)d_in[12];  // (50000,1024)
    const float* out_b   = (const float*)d_in[13];  // (50000,)

    float* out = (float*)d_out;
    float* ws  = (float*)d_ws;

    // 1) embedding + attention + concat
    k_prologue<<<1, 1024, 0, stream>>>(input, hidden, enc, emb_W, attn_W, attn_b,
                                       ws, out);

    // 2) combine GEMV (1024 rows x K=2048) + ReLU  -> x0
    k_gemv16<<<8, 256, 0, stream>>>(comb_W, ws + WS_XA, comb_b, ws + WS_X0,
                                    TWO_H, 64, 1);

    // 3) GRU layer 0
    k_gru_gemv<<<48, 256, 0, stream>>>(gru_Wih, gru_Whh,
                                       gru_bih, gru_bhh,
                                       ws + WS_X0, hidden, ws + WS_G);
    k_gru_gate<<<4, 256, 0, stream>>>(ws + WS_G, hidden, ws + WS_H0, out + OUT_HID);

    // 4) GRU layer 1
    const size_t WOFF = (size_t)THREE_H * H_DIM;
    k_gru_gemv<<<48, 256, 0, stream>>>(gru_Wih + WOFF, gru_Whh + WOFF,
                                       gru_bih + THREE_H, gru_bhh + THREE_H,
                                       ws + WS_H0, hidden + H_DIM, ws + WS_G);
    k_gru_gate<<<4, 256, 0, stream>>>(ws + WS_G, hidden + H_DIM, ws + WS_H1,
                                      out + OUT_HID + H_DIM);

    // 5) output projection (50000 rows x K=1024) -> logits
    k_gemv16<<<391, 256, 0, stream>>>(out_W, ws + WS_H1, out_b, ws + WS_LOGITS,
                                      H_DIM, 3125, 0);

    // 6) log-softmax
    k_lse<<<1, 1024, 0, stream>>>(ws + WS_LOGITS, ws + WS_SCAL);
    k_logsoftmax<<<196, 256, 0, stream>>>(ws + WS_LOGITS, ws + WS_SCAL, out);
}